// ROCKET_84439057039374
// MI455X (gfx1250) — compile-verified
//
#include <hip/hip_runtime.h>
#include <hip/hip_bf16.h>
#include <math.h>

typedef __attribute__((ext_vector_type(2))) float v2f;
typedef __attribute__((ext_vector_type(8))) float v8f;
typedef __attribute__((ext_vector_type(4))) int   v4i;
typedef __attribute__((address_space(1))) v4i     v4i_g;   // global
typedef __attribute__((address_space(3))) v4i     v4i_l;   // LDS

#define T_LEN   2048
#define XP_LEN  2050     // padded signal length (T+2)
#define K_KERN  10000
#define NBINS   2048
#define MAXTILE 4096

#if defined(__has_builtin)
#  if __has_builtin(__builtin_amdgcn_global_load_async_to_lds_b128)
#    define HAVE_ASYNC_LDS 1
#  endif
#endif
#ifndef HAVE_ASYNC_LDS
#  define HAVE_ASYNC_LDS 0
#endif

__device__ __forceinline__ void wait_async0() {
#if defined(__has_builtin) && __has_builtin(__builtin_amdgcn_s_wait_asynccnt)
    __builtin_amdgcn_s_wait_asynccnt(0);
#else
    asm volatile("s_wait_asynccnt 0x0" ::: "memory");
#endif
}

// ---------------------------------------------------------------------------
// Grouping: histogram over dilation, serial scan -> tile list, scatter
// ---------------------------------------------------------------------------
__global__ void rocket_init(int* __restrict__ hist) {
    int i = blockIdx.x * blockDim.x + threadIdx.x;
    if (i < NBINS) hist[i] = 0;
}

__global__ void rocket_hist(const int* __restrict__ dil, int* __restrict__ hist) {
    int i = blockIdx.x * blockDim.x + threadIdx.x;
    if (i < K_KERN) atomicAdd(&hist[min(dil[i], NBINS - 1)], 1);
}

__global__ void rocket_scan(const int* __restrict__ hist, int* __restrict__ cursor,
                            int* __restrict__ tileDil, int* __restrict__ tileBase,
                            int* __restrict__ numTiles) {
    if (threadIdx.x == 0 && blockIdx.x == 0) {
        int base = 0, nt = 0;
        for (int d = 0; d < NBINS; ++d) {
            int cnt = hist[d];
            cursor[d] = base;                       // group start (scatter cursor)
            for (int tb = 0; tb < cnt && nt < MAXTILE; tb += 16) {
                tileDil[nt] = d;
                tileBase[nt] = base + tb;
                ++nt;
            }
            base += cnt;
        }
        numTiles[0] = nt;
    }
}

__global__ void rocket_scatter(const int* __restrict__ dil, int* __restrict__ cursor,
                               int* __restrict__ order) {
    int i = blockIdx.x * blockDim.x + threadIdx.x;
    if (i < K_KERN) {
        int d = min(dil[i], NBINS - 1);
        int pos = atomicAdd(&cursor[d], 1);         // cursor ends at group end
        order[pos] = i;
    }
}

// ---------------------------------------------------------------------------
// WMMA conv: one wave per 16-kernel tile (all rows share the dilation).
// D[m,n] = bias_m + sum_j w[m,j] * xp[u0 + n + dil*j]   via 3x wmma f32 16x16x4
// ---------------------------------------------------------------------------
__global__ __launch_bounds__(32) void rocket_conv_wmma(
    const float* __restrict__ x, const float* __restrict__ w, const float* __restrict__ b,
    const int* __restrict__ pad, const int* __restrict__ Ls,
    const int* __restrict__ order, const int* __restrict__ cursor,
    const int* __restrict__ tileDil, const int* __restrict__ tileBase,
    const int* __restrict__ numTiles, float* __restrict__ feats)
{
    // xbuf+4 is 16B-aligned and receives x[0..2047]; xp = xbuf+3 so xp[1]==xbuf[4].
    __shared__ __align__(16) float xbuf[2056];
    float* xp = xbuf + 3;

    int tile = blockIdx.x;
    if (tile >= numTiles[0]) return;                // uniform exit, EXEC stays full

    int lane = threadIdx.x;                         // wave32
    int d    = tileDil[tile];
    int base = tileBase[tile];
    int gEnd = cursor[d];                           // group end after scatter

#if HAVE_ASYNC_LDS
    // Async DMA global->LDS: 16B per lane per iter, 8KB total, no VGPR round-trip.
    for (int t = 0; t < 16; ++t) {
        int idx = (t * 32 + lane) * 4;              // float index into x
        __builtin_amdgcn_global_load_async_to_lds_b128(
            (v4i_g*)(x + idx),
            (v4i_l*)(&xp[1 + idx]),
            0, 0);
    }
    if (lane == 0) { xp[0] = 0.0f; xp[XP_LEN - 1] = 0.0f; }
    wait_async0();
    __syncthreads();
#else
    for (int i = lane; i < XP_LEN; i += 32)
        xp[i] = (i >= 1 && i <= T_LEN) ? x[i - 1] : 0.0f;
    __syncthreads();
#endif

    int m    = lane & 15;                           // tile row this lane feeds (A/B frag)
    int half = lane >> 4;                           // 0: K=0,1   1: K=2,3
    int row  = base + m;
    int kid  = (row < gEnd) ? order[row] : -1;

    float myB = 0.0f, myInvLen = 0.0f;
    int myULo = 0, myUHi = 0;
    if (kid >= 0) {
        myB = b[kid];
        int p  = pad[kid];
        int L  = Ls[kid];
        int ol = T_LEN + 2 * p - d * (L - 1);
        myULo = 1 - p;
        myUHi = myULo + ol;
        myInvLen = 1.0f / (float)ol;
    }

    // A fragments (16x4 f32): lane (m, K = 4c + 2*half + {x:0, y:1})
    int t0 = half * 2;
    auto ldw = [&](int tap) -> float {
        return (kid >= 0 && tap <= 10) ? w[kid * 11 + tap] : 0.0f;
    };
    v2f a0, a1, a2;
    a0.x = ldw(t0);     a0.y = ldw(t0 + 1);
    a1.x = ldw(t0 + 4); a1.y = ldw(t0 + 5);
    a2.x = ldw(t0 + 8); a2.y = ldw(t0 + 9);         // tap 11 guarded -> 0

    // Per-row bias/validity broadcast to the lanes holding rows r + 8*half in C
    float biasArr[8]; int uLoA[8], uHiA[8];
#pragma unroll
    for (int r = 0; r < 8; ++r) {
        int src = r + half * 8;
        biasArr[r] = __shfl(myB,   src, 32);
        uLoA[r]    = __shfl(myULo, src, 32);
        uHiA[r]    = __shfl(myUHi, src, 32);
    }

    int n   = m;                                    // output column in tile
    int kb0 = (t0) * d, kb1 = (t0 + 4) * d, kb2 = (t0 + 8) * d;

    float mx[8]; int cnt[8];
#pragma unroll
    for (int r = 0; r < 8; ++r) { mx[r] = -__builtin_inff(); cnt[r] = 0; }

    for (int u0 = 0; u0 < XP_LEN; u0 += 16) {
        int un = u0 + n;
        v2f b0, b1, b2;                             // B[kk,n] = xp[un + dil*kk], clamped
        b0.x = xp[min(un + kb0,     XP_LEN - 1)];
        b0.y = xp[min(un + kb0 + d, XP_LEN - 1)];
        b1.x = xp[min(un + kb1,     XP_LEN - 1)];
        b1.y = xp[min(un + kb1 + d, XP_LEN - 1)];
        b2.x = xp[min(un + kb2,     XP_LEN - 1)];
        b2.y = xp[min(un + kb2 + d, XP_LEN - 1)];

        v8f c;
#pragma unroll
        for (int r = 0; r < 8; ++r) c[r] = biasArr[r];
        c = __builtin_amdgcn_wmma_f32_16x16x4_f32(false, a0, false, b0, (short)0, c, false, false);
        c = __builtin_amdgcn_wmma_f32_16x16x4_f32(false, a1, false, b1, (short)0, c, false, false);
        c = __builtin_amdgcn_wmma_f32_16x16x4_f32(false, a2, false, b2, (short)0, c, false, false);

#pragma unroll
        for (int r = 0; r < 8; ++r) {
            float v = c[r];
            bool valid = (un >= uLoA[r]) && (un < uHiA[r]);
            cnt[r] += (valid && v > 0.0f) ? 1 : 0;
            if (valid) mx[r] = fmaxf(mx[r], v);
        }
    }

    // Reduce over the 16 lanes of each half (rows r in lanes<16, r+8 in lanes>=16)
#pragma unroll
    for (int r = 0; r < 8; ++r) {
#pragma unroll
        for (int off = 1; off < 16; off <<= 1) {
            cnt[r] +=       __shfl_xor(cnt[r], off, 32);
            mx[r]  = fmaxf(mx[r], __shfl_xor(mx[r], off, 32));
        }
    }

    // lanes 0-7 write rows 0-7 (r=lane); lanes 24-31 write rows 8-15 (r=lane-24).
    // In both cases this lane's own kid / myInvLen correspond to row (lane&15).
    bool writer = (lane < 8) || (lane >= 24);
    if (writer && kid >= 0) {
        int r = lane & 7;
        feats[kid]          = (float)cnt[r] * myInvLen;   // PPV
        feats[K_KERN + kid] = mx[r];                      // max
    }
}

// ---------------------------------------------------------------------------
// Fallback conv (no workspace for grouping): 1 thread per kernel, xp in LDS
// ---------------------------------------------------------------------------
__global__ __launch_bounds__(256) void rocket_conv_simple(
    const float* __restrict__ x, const float* __restrict__ w, const float* __restrict__ b,
    const int* __restrict__ dil, const int* __restrict__ pad, const int* __restrict__ Ls,
    float* __restrict__ feats)
{
    __shared__ float xp[XP_LEN];
    for (int i = threadIdx.x; i < XP_LEN; i += 256)
        xp[i] = (i >= 1 && i <= T_LEN) ? x[i - 1] : 0.0f;
    __syncthreads();

    int i = blockIdx.x * 256 + threadIdx.x;
    if (i >= K_KERN) return;
    float wt[11];
#pragma unroll
    for (int j = 0; j < 11; ++j) wt[j] = w[i * 11 + j];
    int d = dil[i], p = pad[i], L = Ls[i];
    int start = 1 - p;
    int ol = T_LEN + 2 * p - d * (L - 1);
    float bias = b[i];
    int cnt = 0; float mx = -__builtin_inff();
    for (int s = 0; s < ol; ++s) {
        float acc = bias;
        int bi = start + s;
#pragma unroll
        for (int j = 0; j < 11; ++j) acc += wt[j] * xp[min(bi + d * j, XP_LEN - 1)];
        cnt += (acc > 0.0f) ? 1 : 0;
        mx = fmaxf(mx, acc);
    }
    feats[i]          = (float)cnt / (float)ol;
    feats[K_KERN + i] = mx;
}

// ---------------------------------------------------------------------------
// Classifier: logits = feats @ Wc + bc via WMMA (K chunks of 4), softmax
// ---------------------------------------------------------------------------
__global__ __launch_bounds__(256) void rocket_classifier(
    const float* __restrict__ feats, const float* __restrict__ Wc,
    const float* __restrict__ bc, float* __restrict__ out)
{
    __shared__ float partial[8][16];
    int lane = threadIdx.x & 31;
    int wv   = threadIdx.x >> 5;
    int n    = lane & 15;

    v8f c;
#pragma unroll
    for (int r = 0; r < 8; ++r) c[r] = 0.0f;

    for (int t = wv; t < (2 * K_KERN) / 4; t += 8) {
        int kbase = t * 4;
        float ax = 0.0f, ay = 0.0f;                 // A row 0 only
        if (lane == 0)       { ax = feats[kbase];     ay = feats[kbase + 1]; }
        else if (lane == 16) { ax = feats[kbase + 2]; ay = feats[kbase + 3]; }
        v2f a; a.x = ax; a.y = ay;
        int kk0 = kbase + ((lane < 16) ? 0 : 2);
        v2f bb;
        bb.x = (n < 10) ? Wc[kk0 * 10 + n]       : 0.0f;
        bb.y = (n < 10) ? Wc[(kk0 + 1) * 10 + n] : 0.0f;
        c = __builtin_amdgcn_wmma_f32_16x16x4_f32(false, a, false, bb, (short)0, c, false, false);
    }
    if (lane < 16) partial[wv][lane] = c[0];        // M=0 row lives in c[0], lanes 0-15
    __syncthreads();

    if (threadIdx.x == 0) {
        float logits[10], mxv = -__builtin_inff();
        for (int cc = 0; cc < 10; ++cc) {
            float s = bc[cc];
            for (int wvi = 0; wvi < 8; ++wvi) s += partial[wvi][cc];
            logits[cc] = s;
            mxv = fmaxf(mxv, s);
        }
        float sum = 0.0f;
        for (int cc = 0; cc < 10; ++cc) { logits[cc] = expf(logits[cc] - mxv); sum += logits[cc]; }
        float inv = 1.0f / sum;
        for (int cc = 0; cc < 10; ++cc) out[cc] = logits[cc] * inv;
    }
}

// ---------------------------------------------------------------------------
extern "C" void kernel_launch(void* const* d_in, const int* in_sizes, int n_in,
                              void* d_out, int out_size, void* d_ws, size_t ws_size,
                              hipStream_t stream)
{
    const float* x   = (const float*)d_in[0];
    const float* w   = (const float*)d_in[1];
    const float* b   = (const float*)d_in[2];
    const int*   dil = (const int*)d_in[3];
    const int*   pad = (const int*)d_in[4];
    const int*   Ls  = (const int*)d_in[5];
    const float* Wc  = (const float*)d_in[6];
    const float* bc  = (const float*)d_in[7];
    float* out = (float*)d_out;

    char* ws = (char*)d_ws;
    float* feats   = (float*)(ws);                   //  20000 f32
    int*   hist    = (int*)(ws + 80000);             //   2048 i32
    int*   cursor  = (int*)(ws + 88192);             //   2048 i32
    int*   order   = (int*)(ws + 96384);             //  10000 i32
    int*   tileDil = (int*)(ws + 136384);            //   4096 i32
    int*   tileBase= (int*)(ws + 152768);            //   4096 i32
    int*   numT    = (int*)(ws + 169152);            //      1 i32
    const size_t needed = 169156;

    if (ws_size >= needed) {
        rocket_init   <<<(NBINS + 255) / 256, 256, 0, stream>>>(hist);
        rocket_hist   <<<(K_KERN + 255) / 256, 256, 0, stream>>>(dil, hist);
        rocket_scan   <<<1, 32, 0, stream>>>(hist, cursor, tileDil, tileBase, numT);
        rocket_scatter<<<(K_KERN + 255) / 256, 256, 0, stream>>>(dil, cursor, order);
        rocket_conv_wmma<<<MAXTILE, 32, 0, stream>>>(x, w, b, pad, Ls, order, cursor,
                                                     tileDil, tileBase, numT, feats);
    } else {
        rocket_conv_simple<<<(K_KERN + 255) / 256, 256, 0, stream>>>(x, w, b, dil, pad, Ls, feats);
    }
    rocket_classifier<<<1, 256, 0, stream>>>(feats, Wc, bc, out);
}